// AttentionCropLayer_86431921865103
// MI455X (gfx1250) — compile-verified
//
#include <hip/hip_runtime.h>
#include <math.h>

// ---------------------------------------------------------------------------
// AttentionCropLayer for MI455X (gfx1250).
// Memory-bound gather: roofline ~192 MB traffic => ~8.3 us @ 23.3 TB/s.
// Uses CDNA5 async global->LDS staging (ASYNCcnt) when the builtin exists.
// ---------------------------------------------------------------------------

#define SDIM  448   // source image H/W
#define OUTD  224   // output H/W

#if __has_builtin(__builtin_amdgcn_global_load_async_to_lds_b64) && \
    __has_builtin(__builtin_amdgcn_s_wait_asynccnt)
#define USE_ASYNC_LDS 1
#else
#define USE_ASYNC_LDS 0
#endif

#if USE_ASYNC_LDS
typedef int v2i __attribute__((ext_vector_type(2)));
typedef __attribute__((address_space(1))) v2i* gv2i_p;   // global
typedef __attribute__((address_space(3))) v2i* lv2i_p;   // LDS
#endif

__device__ __forceinline__ float sig10(float z) {
    // sigmoid(10*z), full-precision exp to match the reference closely
    return 1.0f / (1.0f + expf(-10.0f * z));
}

// ---------------------------------------------------------------------------
// Phase 1: per-image tap tables.
// ws layout per image (stride 8*OUTD 32-bit words):
//   [0*OUTD) a0   row weight for tap r0 (includes mx[r0])
//   [1*OUTD) a1   row weight for tap r1 (includes mx[r1])
//   [2*OUTD) b0   col weight for tap c0 (includes my[c0])
//   [3*OUTD) b1   col weight for tap c1 (includes my[c1])
//   [4*OUTD) r0 (int)   [5*OUTD) r1 (int)
//   [6*OUTD) c0 (int)   [7*OUTD) c1 (int)
// ---------------------------------------------------------------------------
__global__ void ac_build_tables(const float* __restrict__ locs,
                                float* __restrict__ ws) {
    const int b = blockIdx.x;
    const int j = threadIdx.x;            // 0..OUTD-1
    const float Sf = (float)SDIM;

    float tx = locs[b * 3 + 0];
    float ty = locs[b * 3 + 1];
    float tl = locs[b * 3 + 2];

    tl = fminf(fmaxf(tl, Sf / 3.0f), Sf * (2.0f / 3.0f));
    tx = fminf(fmaxf(tx, tl), Sf - tl);
    ty = fminf(fmaxf(ty, tl), Sf - tl);

    const float w_off = fmaxf(floorf(tx - tl), 0.0f);
    const float h_off = fmaxf(floorf(ty - tl), 0.0f);
    const float w_end = fminf(floorf(tx + tl), Sf);
    const float h_end = fminf(floorf(ty + tl), Sf);
    const float Lw = w_end - w_off;
    const float Lh = h_end - h_off;

    const float t = (float)j / (float)(OUTD - 1);

    // rows (dim 1 of the image, governed by tx / w_off / w_end)
    const float r   = w_off + t * (Lw - 1.0f);
    const float r0f = fminf(fmaxf(floorf(r), 0.0f), Sf - 1.0f);
    const float wr  = r - r0f;
    const int   r0  = (int)r0f;
    const int   r1  = (r0 + 1 < SDIM) ? (r0 + 1) : (SDIM - 1);
    const float mx0 = sig10(r0f - w_off) - sig10(r0f - w_end);
    const float mx1 = sig10((float)r1 - w_off) - sig10((float)r1 - w_end);

    // cols (dim 2 of the image, governed by ty / h_off / h_end)
    const float c   = h_off + t * (Lh - 1.0f);
    const float c0f = fminf(fmaxf(floorf(c), 0.0f), Sf - 1.0f);
    const float wc  = c - c0f;
    const int   c0  = (int)c0f;
    const int   c1  = (c0 + 1 < SDIM) ? (c0 + 1) : (SDIM - 1);
    const float my0 = sig10(c0f - h_off) - sig10(c0f - h_end);
    const float my1 = sig10((float)c1 - h_off) - sig10((float)c1 - h_end);

    float* base = ws + (size_t)b * (8 * OUTD);
    int*   ibase = (int*)base;
    base[0 * OUTD + j] = (1.0f - wr) * mx0;
    base[1 * OUTD + j] = wr * mx1;
    base[2 * OUTD + j] = (1.0f - wc) * my0;
    base[3 * OUTD + j] = wc * my1;
    ibase[4 * OUTD + j] = r0;
    ibase[5 * OUTD + j] = r1;
    ibase[6 * OUTD + j] = c0;
    ibase[7 * OUTD + j] = c1;
}

// ---------------------------------------------------------------------------
// Phase 2: one block per output row (b, j); 224 threads (7 wave32).
// Stage rows r0/r1 for 3 channels into LDS (async path on gfx1250), then
// each lane does 4 LDS gathers per channel + coalesced stores.
// ---------------------------------------------------------------------------
__global__ __launch_bounds__(OUTD) void ac_crop_upsample(
        const float* __restrict__ img,
        const float* __restrict__ ws,
        float* __restrict__ out) {
    const int bj = blockIdx.x;
    const int b  = bj / OUTD;
    const int j  = bj % OUTD;
    const int k  = threadIdx.x;           // output column, 0..223

    const float* tb = ws + (size_t)b * (8 * OUTD);
    const int*   ti = (const int*)tb;

    const float a0 = tb[0 * OUTD + j];
    const float a1 = tb[1 * OUTD + j];
    const int   r0 = ti[4 * OUTD + j];
    const int   r1 = ti[5 * OUTD + j];

    const float b0 = tb[2 * OUTD + k];
    const float b1 = tb[3 * OUTD + k];
    const int   c0 = ti[6 * OUTD + k];
    const int   c1 = ti[7 * OUTD + k];

    // 3 channels x {r0,r1} rows of 448 floats = 10.5 KB LDS
    __shared__ __align__(16) float rows[3][2][SDIM];

    const float* imgb = img + (size_t)b * 3 * SDIM * SDIM;

#pragma unroll
    for (int ch = 0; ch < 3; ++ch) {
        const float* g0 = imgb + ((size_t)ch * SDIM + (size_t)r0) * SDIM;
        const float* g1 = imgb + ((size_t)ch * SDIM + (size_t)r1) * SDIM;
#if USE_ASYNC_LDS
        // each of 224 lanes moves one b64 (2 floats) per row: 224*2 = 448
        float* g0nc = const_cast<float*>(g0 + 2 * k);
        float* g1nc = const_cast<float*>(g1 + 2 * k);
        __builtin_amdgcn_global_load_async_to_lds_b64(
            (gv2i_p)g0nc, (lv2i_p)&rows[ch][0][2 * k], 0, 0);
        __builtin_amdgcn_global_load_async_to_lds_b64(
            (gv2i_p)g1nc, (lv2i_p)&rows[ch][1][2 * k], 0, 0);
#else
        *(float2*)&rows[ch][0][2 * k] = *(const float2*)(g0 + 2 * k);
        *(float2*)&rows[ch][1][2 * k] = *(const float2*)(g1 + 2 * k);
#endif
    }

#if USE_ASYNC_LDS
    __builtin_amdgcn_s_wait_asynccnt(0);
#endif
    __syncthreads();

#pragma unroll
    for (int ch = 0; ch < 3; ++ch) {
        const float v00 = rows[ch][0][c0];
        const float v01 = rows[ch][0][c1];
        const float v10 = rows[ch][1][c0];
        const float v11 = rows[ch][1][c1];
        const float res = a0 * (b0 * v00 + b1 * v01) +
                          a1 * (b0 * v10 + b1 * v11);
        out[(((size_t)(b * 3 + ch)) * OUTD + (size_t)j) * OUTD + (size_t)k] = res;
    }
}

extern "C" void kernel_launch(void* const* d_in, const int* in_sizes, int n_in,
                              void* d_out, int out_size, void* d_ws, size_t ws_size,
                              hipStream_t stream) {
    const float* images = (const float*)d_in[0];  // [B,3,448,448] f32
    const float* locs   = (const float*)d_in[1];  // [B,3] f32
    float* out = (float*)d_out;                   // [B,3,224,224] f32
    float* ws  = (float*)d_ws;                    // >= B*8*224*4 bytes

    const int B = in_sizes[1] / 3;

    ac_build_tables<<<dim3(B), dim3(OUTD), 0, stream>>>(locs, ws);
    ac_crop_upsample<<<dim3(B * OUTD), dim3(OUTD), 0, stream>>>(images, ws, out);
}